// TimeDomainModel_20555713478789
// MI455X (gfx1250) — compile-verified
//
#include <hip/hip_runtime.h>
#include <math.h>

// Balloon-Windkessel BOLD model: 500-step serial Euler scan + std-normalize.
// Single-wave (wave32) kernel; noise staged via CDNA5 async global->LDS;
// mean/variance reductions done on the matrix pipe (WMMA x ones-matrix).

#define LENGTH 500
#define NPAD   512          // 8 WMMA blocks * 64 elements
#define V0C    0.02f

typedef __attribute__((ext_vector_type(2))) float v2f;
typedef __attribute__((ext_vector_type(8))) float v8f;

__device__ __forceinline__ float wave_sum32(float x) {
#pragma unroll
    for (int o = 16; o > 0; o >>= 1) x += __shfl_xor(x, o, 32);
    return x;
}

__global__ __launch_bounds__(32)
void bold_balloon_kernel(const float* __restrict__ sigma_p,
                         const float* __restrict__ mu_p,
                         const float* __restrict__ lamb_p,
                         const float* __restrict__ beta_p,
                         const float* __restrict__ psi_p,
                         const float* __restrict__ phi_p,
                         const float* __restrict__ chi_p,
                         const float* __restrict__ noise,
                         float* __restrict__ out)
{
    __shared__ float s_eps[NPAD];
    __shared__ float s_y[NPAD];

    const int lane = threadIdx.x;   // 0..31, one wave

    // ---- CDNA5 async global->LDS staging of the noise vector ----
    {
        unsigned long long gbase = (unsigned long long)noise;
        for (int i = lane; i < LENGTH; i += 32) {
            unsigned lds_off = (unsigned)(unsigned long long)&s_eps[i]; // flat LDS addr low 32b == LDS offset
            unsigned goff    = (unsigned)(i * 4);
            asm volatile("global_load_async_to_lds_b32 %0, %1, %2"
                         :: "v"(lds_off), "v"(goff), "s"(gbase)
                         : "memory");
        }
    }
    // zero-pad the output series slots 500..511 (needed by WMMA reduction)
    if (lane < NPAD - LENGTH) s_y[LENGTH + lane] = 0.0f;
    asm volatile("s_wait_asynccnt 0x0" ::: "memory");
    __syncthreads();

    // ---- sequential 500-step Euler integration on lane 0 ----
    if (lane == 0) {
        const float sigma = *sigma_p, mu = *mu_p, lamb = *lamb_p, beta = *beta_p;
        const float psi = *psi_p, phi = *phi_p, chi = *chi_p;
        const float dt = 0.1f;
        const float sqdt = sqrtf(0.1f);
        const float k1 = 7.0f * beta;
        const float k2 = 2.0f;
        const float k3 = 2.0f * beta - 0.2f;
        const float l1mb = logf(1.0f - beta);   // (1-beta)^(1/f) = exp(l1mb/f), base > 0

        float z = 0.0f, s = 0.0f, f = 1.0f, v = 1.0f, q = 1.0f;
        for (int t = 0; t < LENGTH; ++t) {
            const float eps = s_eps[t];
            // neural state (cubic Hopf) + injected noise
            const float dz = sigma * z - mu * z * z * z + lamb;
            const float zn = z + dt * dz + sqdt * eps;
            // hemodynamics (alpha = 1 -> v^(1/alpha) == v)
            const float fv  = v;
            const float ds_ = z - phi * s - psi * (f - 1.0f);
            const float df_ = s;
            const float dv_ = (f - fv) / chi;
            const float E   = 1.0f - expf(l1mb / f);
            const float dq_ = (f * E / beta - fv * q / v) / chi;
            s += dt * ds_;
            f += dt * df_;
            v += dt * dv_;
            q += dt * dq_;
            z  = zn;
            const float y = V0C * (k1 * (1.0f - q) + k2 * (1.0f - q / v) + k3 * (1.0f - v));
            s_y[t] = y;
        }
    }
    __syncthreads();   // s_y visible to all lanes; EXEC reconverged (WMMA needs all-ones EXEC)

    v2f ones; ones.x = 1.0f; ones.y = 1.0f;

    // ---- pass 1: sum(y) via WMMA (A x ones): every row-sum replicated 16x ----
    v8f acc = {0.f, 0.f, 0.f, 0.f, 0.f, 0.f, 0.f, 0.f};
#pragma unroll
    for (int blk = 0; blk < 8; ++blk) {
        const int b = blk * 64 + lane * 2;
        v2f a; a.x = s_y[b]; a.y = s_y[b + 1];   // pad slots are 0 -> no contribution
        acc = __builtin_amdgcn_wmma_f32_16x16x4_f32(false, a, false, ones,
                                                    (short)0, acc, false, false);
    }
    float p = acc[0] + acc[1] + acc[2] + acc[3] + acc[4] + acc[5] + acc[6] + acc[7];
    const float mean = wave_sum32(p) * (1.0f / 16.0f) * (1.0f / (float)LENGTH);

    // ---- pass 2: sum((y-mean)^2) via WMMA; mask out pad slots ----
    v8f acc2 = {0.f, 0.f, 0.f, 0.f, 0.f, 0.f, 0.f, 0.f};
#pragma unroll
    for (int blk = 0; blk < 8; ++blk) {
        const int b = blk * 64 + lane * 2;
        const float d0 = (b     < LENGTH) ? (s_y[b]     - mean) : 0.0f;
        const float d1 = (b + 1 < LENGTH) ? (s_y[b + 1] - mean) : 0.0f;
        v2f a; a.x = d0 * d0; a.y = d1 * d1;
        acc2 = __builtin_amdgcn_wmma_f32_16x16x4_f32(false, a, false, ones,
                                                     (short)0, acc2, false, false);
    }
    float p2 = acc2[0] + acc2[1] + acc2[2] + acc2[3] + acc2[4] + acc2[5] + acc2[6] + acc2[7];
    const float var = wave_sum32(p2) * (1.0f / 16.0f) * (1.0f / (float)(LENGTH - 1));
    const float inv = 1.0f / sqrtf(var);   // unbiased std (ddof = 1)

    // ---- normalize and store ----
    for (int i = lane; i < LENGTH; i += 32) out[i] = s_y[i] * inv;
}

extern "C" void kernel_launch(void* const* d_in, const int* in_sizes, int n_in,
                              void* d_out, int out_size, void* d_ws, size_t ws_size,
                              hipStream_t stream) {
    (void)in_sizes; (void)n_in; (void)out_size; (void)d_ws; (void)ws_size;
    bold_balloon_kernel<<<1, 32, 0, stream>>>(
        (const float*)d_in[0],   // sigma
        (const float*)d_in[1],   // mu
        (const float*)d_in[2],   // lamb
        (const float*)d_in[3],   // beta
        (const float*)d_in[4],   // psi
        (const float*)d_in[5],   // phi
        (const float*)d_in[6],   // chi
        (const float*)d_in[7],   // noise [500]
        (float*)d_out);          // y / std(y) [500]
}